// LGE_Exp_77601469104294
// MI455X (gfx1250) — compile-verified
//
#include <hip/hip_runtime.h>
#include <math.h>

#define LGE_NF   4
#define LGE_NDIR 4
#define LGE_VOL  (16*16*16*32)   // 131072 sites

// reciprocals 1/k for the Horner Taylor loop (avoid fp64 divides)
__constant__ double c_invk[15] = {
    0.0, 1.0, 1.0/2.0, 1.0/3.0, 1.0/4.0, 1.0/5.0, 1.0/6.0, 1.0/7.0,
    1.0/8.0, 1.0/9.0, 1.0/10.0, 1.0/11.0, 1.0/12.0, 1.0/13.0, 1.0/14.0 };

struct cplx { double re, im; };

__device__ __forceinline__ cplx cmul(const cplx a, const cplx b) {
    cplx r;
    r.re = fma(a.re, b.re, -(a.im * b.im));
    r.im = fma(a.re, b.im,   a.im * b.re);
    return r;
}
__device__ __forceinline__ cplx cfma(const cplx a, const cplx b, cplx c) {
    c.re = fma(a.re, b.re, fma(-a.im, b.im, c.re));
    c.im = fma(a.re, b.im, fma( a.im, b.re, c.im));
    return c;
}

// C = A * B, 3x3 complex (27 complex FMA = 108 fp64 FMA)
__device__ __forceinline__ void mm33(const cplx* A, const cplx* B, cplx* __restrict__ C) {
#pragma unroll
    for (int j = 0; j < 3; ++j) {
#pragma unroll
        for (int k = 0; k < 3; ++k) {
            cplx acc = cmul(A[j*3+0], B[0*3+k]);
            acc = cfma(A[j*3+1], B[1*3+k], acc);
            acc = cfma(A[j*3+2], B[2*3+k], acc);
            C[j*3+k] = acc;
        }
    }
}

__global__ __launch_bounds__(256)
void lge_exp_mul_kernel(const double*  __restrict__ wts,   // [4][4] complex
                        const double2* __restrict__ W,     // [4][VOL][3][3] complex
                        const double2* __restrict__ U,     // [4][VOL][3][3] complex
                        double2*       __restrict__ out)   // [4][VOL][3][3] complex
{
    __shared__ double2 ldsU[256 * 9];          // 36 KB: one 3x3 complex U tile per lane

    const int       tid = threadIdx.x;
    const long long gt  = (long long)blockIdx.x * 256 + tid;
    const int       u   = (int)(gt & 3);       // 4 adjacent lanes share a site -> W loads broadcast
    const long long s   = gt >> 2;

    // ---- CDNA5 async stage: copy this lane's U tile (144 B) global -> LDS ----
    // Overlaps the HBM load with the ~2000-cycle fp64 expm below and keeps 36
    // VGPRs free during the register-heavy part. Tracked by ASYNCcnt.
    {
        const unsigned lbase = (unsigned)(size_t)(&ldsU[tid * 9]);
        const unsigned long long ga =
            (unsigned long long)(const void*)(U + ((long long)u * LGE_VOL + s) * 9);
        // INST_OFFSET is added to both the LDS and the global address (ISA 15.18.3)
#define LGE_ASYNC_CP(OFF) \
        asm volatile("global_load_async_to_lds_b128 %0, %1, off offset:" #OFF \
                     :: "v"(lbase), "v"(ga) : "memory")
        LGE_ASYNC_CP(0);  LGE_ASYNC_CP(16); LGE_ASYNC_CP(32);
        LGE_ASYNC_CP(48); LGE_ASYNC_CP(64); LGE_ASYNC_CP(80);
        LGE_ASYNC_CP(96); LGE_ASYNC_CP(112); LGE_ASYNC_CP(128);
#undef LGE_ASYNC_CP
    }

    // ---- build A = -0.5 * sum_i w[u][i] * (W_i^dag - W_i - tr(W_i^dag - W_i)/3 * I) ----
    cplx A[9];
#pragma unroll
    for (int e = 0; e < 9; ++e) A[e] = {0.0, 0.0};

#pragma unroll
    for (int i = 0; i < LGE_NF; ++i) {
        const double2* Wp = W + ((long long)i * LGE_VOL + s) * 9;
        double2 wm[9];
#pragma unroll
        for (int e = 0; e < 9; ++e) wm[e] = Wp[e];

        // tr(W^dag - W) = -2i * Im(tr W)
        const double trim = wm[0].y + wm[4].y + wm[8].y;
        const double trd_im = -2.0 * trim / 3.0;

        cplx c;                                  // -0.5 * weights[u][i]
        c.re = -0.5 * wts[(u * LGE_NF + i) * 2 + 0];
        c.im = -0.5 * wts[(u * LGE_NF + i) * 2 + 1];

#pragma unroll
        for (int j = 0; j < 3; ++j) {
#pragma unroll
            for (int k = 0; k < 3; ++k) {
                cplx h;                          // conj(W[k][j]) - W[j][k]
                h.re =  wm[k*3+j].x - wm[j*3+k].x;
                h.im = -wm[k*3+j].y - wm[j*3+k].y;
                if (j == k) h.im -= trd_im;
                A[j*3+k] = cfma(c, h, A[j*3+k]);
            }
        }
    }

    // ---- expm(A): scaling (||A||_inf <= 0.5) + degree-14 Taylor Horner + squaring ----
    double nrm = 0.0;
#pragma unroll
    for (int j = 0; j < 3; ++j) {
        double r = 0.0;
#pragma unroll
        for (int k = 0; k < 3; ++k)
            r += fabs(A[j*3+k].re) + fabs(A[j*3+k].im);
        nrm = fmax(nrm, r);
    }
    int sc = 0;
    if (nrm > 0.5) {
        const long long bits = __double_as_longlong(nrm);
        const int e = (int)((bits >> 52) & 0x7FF) - 1023;   // floor(log2(nrm))
        sc = e + 2;                                         // nrm * 2^-sc <= 0.5
        if (sc < 1)  sc = 1;
        if (sc > 60) sc = 60;
        const double scl = __longlong_as_double((long long)(1023 - sc) << 52); // 2^-sc
#pragma unroll
        for (int q = 0; q < 9; ++q) { A[q].re *= scl; A[q].im *= scl; }
    }

    cplx P[9], T[9];
#pragma unroll
    for (int e = 0; e < 9; ++e) P[e] = {0.0, 0.0};
    P[0].re = P[4].re = P[8].re = 1.0;

    for (int k = 14; k >= 1; --k) {              // P = I + (A/k) * P  (Horner)
        mm33(A, P, T);
        const double inv = c_invk[k];
#pragma unroll
        for (int e = 0; e < 9; ++e) { P[e].re = T[e].re * inv; P[e].im = T[e].im * inv; }
        P[0].re += 1.0; P[4].re += 1.0; P[8].re += 1.0;
    }
    for (int q = 0; q < sc; ++q) {               // undo scaling: P <- P^2, sc times
        mm33(P, P, T);
#pragma unroll
        for (int e = 0; e < 9; ++e) P[e] = T[e];
    }

    // ---- out = expm(A) @ U  (U arrives via async LDS staging) ----
    asm volatile("s_wait_asynccnt 0" ::: "memory");
    cplx Um[9];
#pragma unroll
    for (int e = 0; e < 9; ++e) {
        const double2 v = ldsU[tid * 9 + e];
        Um[e].re = v.x; Um[e].im = v.y;
    }
    mm33(P, Um, T);

    double2* op = out + ((long long)u * LGE_VOL + s) * 9;
#pragma unroll
    for (int e = 0; e < 9; ++e) {
        double2 v; v.x = T[e].re; v.y = T[e].im;
        op[e] = v;
    }
}

extern "C" void kernel_launch(void* const* d_in, const int* in_sizes, int n_in,
                              void* d_out, int out_size, void* d_ws, size_t ws_size,
                              hipStream_t stream) {
    (void)in_sizes; (void)n_in; (void)out_size; (void)d_ws; (void)ws_size;
    const double*  wts = (const double*)d_in[0];   // complex128 -> interleaved double
    const double2* W   = (const double2*)d_in[1];
    const double2* U   = (const double2*)d_in[2];
    double2*       out = (double2*)d_out;

    const long long total  = (long long)LGE_NDIR * LGE_VOL;   // 524288 threads
    const int       block  = 256;                              // 8 wave32 waves
    const int       blocks = (int)(total / block);             // 2048

    hipLaunchKernelGGL(lge_exp_mul_kernel, dim3(blocks), dim3(block), 0, stream,
                       wts, W, U, out);
}